// ExperienceMemory_51874615001332
// MI455X (gfx1250) — compile-verified
//
#include <hip/hip_runtime.h>
#include <math.h>

typedef __attribute__((ext_vector_type(2))) float v2f;
typedef __attribute__((ext_vector_type(8))) float v8f;

constexpr int kB = 8, kS = 2048, kH = 1024, kM = 100000, kPD = 128, kSD = 128;
constexpr int kTopK = 5;
constexpr int kSChunks = 16, kSPer = kS / kSChunks;   // 16 chunks of 128

// workspace layout (float offsets)
constexpr int WS_PARTIAL = 0;                               // B*16*H   = 131072
constexpr int WS_MEAN    = WS_PARTIAL + kB * kSChunks * kH; // B*H      =   8192
constexpr int WS_CP      = WS_MEAN + kB * kH;               // 16*PD    =   2048 (rows 8..15 zero)
constexpr int WS_SCORES  = WS_CP + 16 * kPD;                // B*M      = 800000
constexpr int WS_E       = WS_SCORES + kB * kM;             // B*H      =   8192

// ---------------- Stage 1a: partial sums of x over S chunks ----------------
__global__ void em_partial_mean(const float* __restrict__ x, float* __restrict__ partial) {
    const int hc = blockIdx.x & 3;          // 4 chunks of 256 h
    const int sc = (blockIdx.x >> 2) & 15;  // 16 chunks of 128 s
    const int b  = blockIdx.x >> 6;
    const int h  = hc * 256 + threadIdx.x;
    const float* base = x + ((size_t)b * kS + (size_t)sc * kSPer) * kH + h;
    float s = 0.f;
    for (int i = 0; i < kSPer; ++i) s += base[(size_t)i * kH];
    partial[((size_t)b * kSChunks + sc) * kH + h] = s;
}

// ---------------- Stage 1b: finish mean ----------------
__global__ void em_finish_mean(const float* __restrict__ partial, float* __restrict__ mean) {
    const int idx = blockIdx.x * 256 + threadIdx.x;   // B*H
    const int b = idx >> 10, h = idx & 1023;
    float s = 0.f;
    for (int c = 0; c < kSChunks; ++c) s += partial[((size_t)b * kSChunks + c) * kH + h];
    mean[idx] = s * (1.0f / (float)kS);
}

// ---------------- Stage 2: current_problem = mean @ W_prob + b_prob (padded to 16 rows) ---
__global__ void em_proj_problem(const float* __restrict__ mean, const float* __restrict__ W_prob,
                                const float* __restrict__ b_prob, float* __restrict__ cp) {
    const int o = blockIdx.x * 256 + threadIdx.x;     // 16*128
    const int r = o >> 7, p = o & 127;
    float acc = 0.f;
    if (r < kB) {
        acc = b_prob[p];
        for (int h = 0; h < kH; ++h) acc += mean[r * kH + h] * W_prob[h * kPD + p];
    }
    cp[o] = acc;   // rows 8..15 written as zero -> WMMA A-pad is clean
}

// ---------------- Stage 3: sim GEMM via V_WMMA_F32_16X16X4_F32 + score boosts ------------
// One wave handles a 16-column tile of problem_memory. A = cp (16x128, rows 8..15 zero).
__global__ void em_sim_scores_wmma(const float* __restrict__ cp,   // 16 x 128
                                   const float* __restrict__ pm,   // M x 128
                                   const float* __restrict__ conf,
                                   const float* __restrict__ usage,
                                   const float* __restrict__ success,
                                   float* __restrict__ scores) {   // B x M
    const int wave = (blockIdx.x * blockDim.x + threadIdx.x) >> 5;
    if (wave >= kM / 16) return;            // wave-uniform exit: active waves keep EXEC all-1s
    const int lane = threadIdx.x & 31;
    const int n0   = wave * 16;
    const int m16  = lane & 15;
    const int koff = (lane >> 4) * 2;       // lanes 0-15: K {k0,k0+1}; lanes 16-31: K {k0+2,k0+3}

    const float* arow = cp + m16 * kPD + koff;
    const float* brow = pm + (size_t)(n0 + m16) * kPD + koff;

    v8f acc = {};
#pragma unroll
    for (int k0 = 0; k0 < kPD; k0 += 4) {
        v2f a = *(const v2f*)(arow + k0);
        v2f b = *(const v2f*)(brow + k0);
        acc = __builtin_amdgcn_wmma_f32_16x16x4_f32(false, a, false, b, (short)0, acc,
                                                    false, false);
    }
    // C/D layout: lanes 0-15 / VGPR r -> D[M=r][N=lane]; batch rows are M=0..7
    if (lane < 16) {
        const int n = n0 + lane;
        const float u = usage[n];
        const float boost = 0.1f * logf(u + 1.0f) + 0.2f * conf[n]
                          + 0.3f * (success[n] / (u + 1e-8f));
#pragma unroll
        for (int r = 0; r < kB; ++r)
            scores[(size_t)r * kM + n] = acc[r] + boost;
    }
}

// ---------------- Stage 4: per-batch top-5 + softmax + gather + W_out projection ---------
__global__ void em_topk_combine(const float* __restrict__ scores,
                                const float* __restrict__ sol,     // M x 128
                                const float* __restrict__ W_out,   // 128 x H
                                const float* __restrict__ b_out,
                                float* __restrict__ e) {           // B x H
    const int b = blockIdx.x;
    const int tid = threadIdx.x;
    __shared__ float sv[256 * kTopK];
    __shared__ int   si[256 * kTopK];
    __shared__ float g_attn[kTopK];
    __shared__ int   g_idx[kTopK];
    __shared__ float comb[kSD];

    float bv[kTopK]; int bi[kTopK];
    for (int k = 0; k < kTopK; ++k) { bv[k] = -3.4e38f; bi[k] = 0x7fffffff; }
    for (int n = tid; n < kM; n += 256) {
        const float v = scores[(size_t)b * kM + n];
        for (int k = 0; k < kTopK; ++k) {
            if (v > bv[k] || (v == bv[k] && n < bi[k])) {
                for (int j = kTopK - 1; j > k; --j) { bv[j] = bv[j - 1]; bi[j] = bi[j - 1]; }
                bv[k] = v; bi[k] = n;
                break;
            }
        }
    }
    for (int k = 0; k < kTopK; ++k) { sv[tid * kTopK + k] = bv[k]; si[tid * kTopK + k] = bi[k]; }
    __syncthreads();

    if (tid == 0) {
        float gv[kTopK];
        for (int k = 0; k < kTopK; ++k) {
            float best = -3.4e38f; int besti = 0x7fffffff; int slot = 0;
            for (int s = 0; s < 256 * kTopK; ++s) {
                const float v = sv[s]; const int i = si[s];
                if (v > best || (v == best && i < besti)) { best = v; besti = i; slot = s; }
            }
            gv[k] = best; g_idx[k] = besti; sv[slot] = -3.4e38f;
        }
        const float inv = 1.0f / sqrtf((float)kSD);
        const float mx = gv[0];
        float ex[kTopK]; float sum = 0.f;
        for (int k = 0; k < kTopK; ++k) { ex[k] = expf((gv[k] - mx) * inv); sum += ex[k]; }
        for (int k = 0; k < kTopK; ++k) g_attn[k] = ex[k] / sum;
    }
    __syncthreads();

    if (tid < kSD) {
        float c = 0.f;
        for (int k = 0; k < kTopK; ++k) c += g_attn[k] * sol[(size_t)g_idx[k] * kSD + tid];
        comb[tid] = c;
    }
    __syncthreads();

#pragma unroll
    for (int j = 0; j < kH / 256; ++j) {
        const int h = tid + j * 256;
        float acc = b_out[h];
        for (int d = 0; d < kSD; ++d) acc += comb[d] * W_out[d * kH + h];
        e[(size_t)b * kH + h] = acc;
    }
}

// ---------------- Stage 5: gate[b,s] = sigmoid(x[b,s,:].e[b,:]); gated blend -------------
__global__ void em_gate_blend(const float* __restrict__ x, const float* __restrict__ e,
                              float* __restrict__ out) {
    __shared__ float se[kH];
    const int rowBase = blockIdx.x * 8;           // 8 waves, one row each; same b per block
    const int b = rowBase >> 11;                  // 2048 rows per batch
    for (int i = threadIdx.x; i < kH; i += 256) se[i] = e[(size_t)b * kH + i];
    __syncthreads();

    const int wave = threadIdx.x >> 5;
    const int lane = threadIdx.x & 31;
    const size_t row = (size_t)rowBase + wave;
    const float* xr = x + row * kH;

    float partial = 0.f;
#pragma unroll
    for (int i = 0; i < kH / 32; ++i) {
        const int idx = lane + i * 32;
        partial += xr[idx] * se[idx];
    }
    for (int off = 16; off > 0; off >>= 1)
        partial += __shfl_xor(partial, off, 32);   // wave32 reduction

    const float g = 1.0f / (1.0f + expf(-partial));
    float* orow = out + row * kH;
#pragma unroll
    for (int i = 0; i < kH / 32; ++i) {
        const int idx = lane + i * 32;
        orow[idx] = g * se[idx] + (1.0f - g) * xr[idx];
    }
}

extern "C" void kernel_launch(void* const* d_in, const int* in_sizes, int n_in,
                              void* d_out, int out_size, void* d_ws, size_t ws_size,
                              hipStream_t stream) {
    const float* x       = (const float*)d_in[0];
    const float* pm      = (const float*)d_in[1];
    const float* sol     = (const float*)d_in[2];
    const float* conf    = (const float*)d_in[3];
    const float* W_prob  = (const float*)d_in[4];
    const float* b_prob  = (const float*)d_in[5];
    const float* W_out   = (const float*)d_in[6];
    const float* b_out   = (const float*)d_in[7];
    const float* usage   = (const float*)d_in[8];
    const float* success = (const float*)d_in[9];
    float* out = (float*)d_out;
    float* ws  = (float*)d_ws;

    float* partial = ws + WS_PARTIAL;
    float* mean    = ws + WS_MEAN;
    float* cp      = ws + WS_CP;
    float* scores  = ws + WS_SCORES;
    float* e       = ws + WS_E;

    em_partial_mean<<<kB * kSChunks * (kH / 256), 256, 0, stream>>>(x, partial);
    em_finish_mean<<<(kB * kH) / 256, 256, 0, stream>>>(partial, mean);
    em_proj_problem<<<(16 * kPD) / 256, 256, 0, stream>>>(mean, W_prob, b_prob, cp);

    const int tiles = kM / 16;                       // 6250
    em_sim_scores_wmma<<<(tiles + 7) / 8, 256, 0, stream>>>(cp, pm, conf, usage, success, scores);

    em_topk_combine<<<kB, 256, 0, stream>>>(scores, sol, W_out, b_out, e);
    em_gate_blend<<<(kB * kS) / 8, 256, 0, stream>>>(x, e, out);
}